// SP_conv_67946382622886
// MI455X (gfx1250) — compile-verified
//
#include <hip/hip_runtime.h>
#include <hip/hip_bf16.h>

typedef float v2f __attribute__((ext_vector_type(2)));
typedef float v8f __attribute__((ext_vector_type(8)));

#define Bn 4
#define Cc 64
#define Hh 256
#define Ww 256
#define HW (Hh * Ww)

// ---------------------------------------------------------------------------
// K1: y[b,o,h,w] = sum_c conv_w[o,c] * x[b,c,h,w] + conv_b[o]
// GEMM per batch: Y(64 x 65536) = W(64x64) @ X(64x65536) + b
// Each wave: 64 out-channels x 16 pixels, via 4 m-tiles * 16 k-steps of
// V_WMMA_F32_16X16X4_F32.  Grid exactly covers all tiles -> EXEC all ones.
// ---------------------------------------------------------------------------
__global__ __launch_bounds__(256) void k1_conv1x1_wmma(
    const float* __restrict__ x,     // [B,C,H,W]
    const float* __restrict__ wmat,  // [64,64] row-major (o,c)
    const float* __restrict__ bias,  // [64]
    float* __restrict__ y)           // [B,C,H,W]
{
    const int lane  = threadIdx.x & 31;
    const int gwave = blockIdx.x * 8 + (threadIdx.x >> 5);
    const int tilesPerBatch = HW / 16;            // 4096
    const int b    = gwave / tilesPerBatch;
    const int tile = gwave % tilesPerBatch;
    const int n    = lane & 15;
    const bool hi  = lane >= 16;

    const long xbase = (long)b * Cc * HW + (long)tile * 16 + n;

    v8f acc[4];
#pragma unroll
    for (int mt = 0; mt < 4; ++mt) acc[mt] = (v8f){0, 0, 0, 0, 0, 0, 0, 0};

#pragma unroll 4
    for (int kt = 0; kt < 16; ++kt) {
        const int k0 = kt * 4 + (hi ? 2 : 0);
        // B tile (4x16 of X): lane holds column n, K rows k0 / k0+1
        v2f bm;
        bm.x = x[xbase + (long)k0 * HW];
        bm.y = x[xbase + (long)(k0 + 1) * HW];
#pragma unroll
        for (int mt = 0; mt < 4; ++mt) {
            // A tile (16x4 of conv_w): lane holds row m = mt*16 + (lane&15),
            // K cols k0 / k0+1 (ISA Sec 7.12.2 32-bit A layout). 8B aligned.
            const int m = mt * 16 + n;
            v2f am = *(const v2f*)(wmat + m * 64 + k0);
            acc[mt] = __builtin_amdgcn_wmma_f32_16x16x4_f32(
                false, am, false, bm, (short)0, acc[mt], false, false);
        }
    }

    // D layout: vgpr r, lane -> M = mt*16 + r + (hi?8:0), N = lane&15
    const long ybase = (long)b * Cc * HW + (long)tile * 16 + n;
#pragma unroll
    for (int mt = 0; mt < 4; ++mt) {
#pragma unroll
        for (int r = 0; r < 8; ++r) {
            const int m = mt * 16 + r + (hi ? 8 : 0);
            y[ybase + (long)m * HW] = acc[mt][r] + bias[m];
        }
    }
}

// ---------------------------------------------------------------------------
// K2a: ph[b,c,w] = mean_h y[b,c,h,w]   (thread per (b,c,w), coalesced in w)
// ---------------------------------------------------------------------------
__global__ __launch_bounds__(256) void k2_pool_h(const float* __restrict__ y,
                                                 float* __restrict__ ph)
{
    const int idx = blockIdx.x * 256 + threadIdx.x;   // b*Cc*Ww + c*Ww + w
    const int w  = idx % Ww;
    const int bc = idx / Ww;
    const float* p = y + (long)bc * HW + w;
    float s = 0.f;
    for (int h = 0; h < Hh; ++h) s += p[(long)h * Ww];
    ph[idx] = s * (1.0f / Hh);
}

// ---------------------------------------------------------------------------
// K2b: pv[b,c,h] = mean_w y[b,c,h,w]   (one wave per row, shfl reduction)
// ---------------------------------------------------------------------------
__global__ __launch_bounds__(256) void k2_pool_v(const float* __restrict__ y,
                                                 float* __restrict__ pv)
{
    const int lane = threadIdx.x & 31;
    const int wave = (blockIdx.x * 256 + threadIdx.x) >> 5;  // b*Cc*Hh + c*Hh + h
    const float* p = y + (long)wave * Ww;
    float s = 0.f;
    for (int w = lane; w < Ww; w += 32) s += p[w];
#pragma unroll
    for (int off = 16; off >= 1; off >>= 1) s += __shfl_xor(s, off, 32);
    if (lane == 0) pv[wave] = s * (1.0f / Ww);
}

// ---------------------------------------------------------------------------
// K3: 1D 3-tap conv over strips (used for both (1,3) on ph and (3,1) on pv;
// both weights are flat [o, c, 3]).  out[b,o,t] = sum_{c,j} wt[o,c,j]*in[b,c,t+j-1]
// ---------------------------------------------------------------------------
__global__ __launch_bounds__(256) void k3_strip(const float* __restrict__ pool,
                                                const float* __restrict__ wt,
                                                float* __restrict__ out)
{
    const int idx = blockIdx.x * 256 + threadIdx.x;   // b*Cc*256 + o*256 + t
    const int t = idx % 256;
    const int o = (idx / 256) % Cc;
    const int b = idx / (256 * Cc);
    const float* pb = pool + (long)b * Cc * 256;
    float s = 0.f;
    for (int c = 0; c < Cc; ++c) {
        const float* pc = pb + c * 256;
        const float l = (t > 0)   ? pc[t - 1] : 0.f;
        const float m = pc[t];
        const float r = (t < 255) ? pc[t + 1] : 0.f;
        const float* wv = wt + (o * Cc + c) * 3;
        s += wv[0] * l + wv[1] * m + wv[2] * r;
    }
    out[idx] = s;
}

// ---------------------------------------------------------------------------
// K4: kern[b,q,h,w] = tanh( sum_c fus_w[q,c] * leaky(xh[b,c,w]+xv[b,c,h]) + fus_b[q] )
// ---------------------------------------------------------------------------
__global__ __launch_bounds__(256) void k4_kern(const float* __restrict__ xh,
                                               const float* __restrict__ xv,
                                               const float* __restrict__ fw,  // [9,64]
                                               const float* __restrict__ fb,  // [9]
                                               float* __restrict__ kern)      // [B,9,H,W]
{
    const int idx = blockIdx.x * 256 + threadIdx.x;   // b*HW + h*Ww + w
    const int w = idx % Ww;
    const int h = (idx / Ww) % Hh;
    const int b = idx / HW;
    const float* xhb = xh + (long)b * Cc * Ww + w;    // stride Ww over c
    const float* xvb = xv + (long)b * Cc * Hh + h;    // stride Hh over c
    float acc[9];
#pragma unroll
    for (int q = 0; q < 9; ++q) acc[q] = fb[q];
    for (int c = 0; c < Cc; ++c) {
        const float s = xhb[c * Ww] + xvb[c * Hh];
        const float a = (s > 0.f) ? s : 0.2f * s;
#pragma unroll
        for (int q = 0; q < 9; ++q) acc[q] += fw[q * Cc + c] * a;
    }
    float* kb = kern + (long)b * 9 * HW + (long)h * Ww + w;
#pragma unroll
    for (int q = 0; q < 9; ++q) kb[(long)q * HW] = tanhf(acc[q]);
}

// ---------------------------------------------------------------------------
// K5: out[b,c,h,w] = sum_{i,j} y[b,c,h+i-1,w+j-1] * kern[b,i*3+j,h,w]  (zero pad)
// ---------------------------------------------------------------------------
__global__ __launch_bounds__(256) void k5_ppconv(const float* __restrict__ y,
                                                 const float* __restrict__ kern,
                                                 float* __restrict__ out)
{
    const long idx = (long)blockIdx.x * 256 + threadIdx.x;  // b,c,h,w
    const int w = (int)(idx % Ww);
    const int h = (int)((idx / Ww) % Hh);
    const int c = (int)((idx / HW) % Cc);
    const int b = (int)(idx / ((long)Cc * HW));
    const float* yb = y + ((long)(b * Cc + c)) * HW;
    const float* kb = kern + (long)b * 9 * HW + (long)h * Ww + w;
    float s = 0.f;
#pragma unroll
    for (int i = 0; i < 3; ++i) {
        const int hh = h + i - 1;
        if (hh < 0 || hh >= Hh) continue;
#pragma unroll
        for (int j = 0; j < 3; ++j) {
            const int ww = w + j - 1;
            if (ww < 0 || ww >= Ww) continue;
            s += yb[(long)hh * Ww + ww] * kb[(long)(i * 3 + j) * HW];
        }
    }
    out[idx] = s;
}

// ---------------------------------------------------------------------------
extern "C" void kernel_launch(void* const* d_in, const int* in_sizes, int n_in,
                              void* d_out, int out_size, void* d_ws, size_t ws_size,
                              hipStream_t stream)
{
    const float* x       = (const float*)d_in[0];  // [4,64,256,256]
    const float* conv_w  = (const float*)d_in[1];  // [64,64]
    const float* conv_b  = (const float*)d_in[2];  // [64]
    const float* convh_w = (const float*)d_in[3];  // [64,64,1,3]
    const float* convv_w = (const float*)d_in[4];  // [64,64,3,1]
    const float* fus_w   = (const float*)d_in[5];  // [9,64]
    const float* fus_b   = (const float*)d_in[6];  // [9]
    float* out = (float*)d_out;

    // workspace layout (floats)
    float* ws   = (float*)d_ws;
    float* y    = ws;                               // 16,777,216
    float* kern = y + (long)Bn * Cc * HW;           //  2,359,296
    float* ph   = kern + (long)Bn * 9 * HW;         //     65,536
    float* pv   = ph + Bn * Cc * Ww;                //     65,536
    float* xh   = pv + Bn * Cc * Hh;                //     65,536
    float* xv   = xh + Bn * Cc * Ww;                //     65,536

    // K1: 16384 waves = 2048 blocks of 8 waves
    k1_conv1x1_wmma<<<2048, 256, 0, stream>>>(x, conv_w, conv_b, y);
    // K2a: Bn*Cc*Ww = 65536 threads
    k2_pool_h<<<256, 256, 0, stream>>>(y, ph);
    // K2b: Bn*Cc*Hh = 65536 waves = 8192 blocks
    k2_pool_v<<<8192, 256, 0, stream>>>(y, pv);
    // K3: strips
    k3_strip<<<256, 256, 0, stream>>>(ph, convh_w, xh);
    k3_strip<<<256, 256, 0, stream>>>(pv, convv_w, xv);
    // K4: Bn*HW = 262144 pixels
    k4_kern<<<1024, 256, 0, stream>>>(xh, xv, fus_w, fus_b, kern);
    // K5: Bn*Cc*HW = 16,777,216 outputs
    k5_ppconv<<<65536, 256, 0, stream>>>(y, kern, out);
}